// SuperPointMatchesGenerator_56925496541369
// MI455X (gfx1250) — compile-verified
//
#include <hip/hip_runtime.h>

typedef __attribute__((ext_vector_type(2))) float v2f;
typedef __attribute__((ext_vector_type(8))) float v8f;

#define FLT_MAX_C 3.402823466e+38f

__device__ __forceinline__ unsigned long long umin64(unsigned long long a,
                                                     unsigned long long b) {
  return a < b ? a : b;
}

__device__ __forceinline__ unsigned long long shflxor64(unsigned long long v, int m) {
  unsigned lo = (unsigned)v;
  unsigned hi = (unsigned)(v >> 32);
  lo = (unsigned)__shfl_xor((int)lo, m, 32);
  hi = (unsigned)__shfl_xor((int)hi, m, 32);
  return ((unsigned long long)hi << 32) | lo;
}

// Kernel 1: reproject kpts0 through homography; init col_key to +inf keys and
// the gt_matches1 output section to -1 (d_out is poisoned by the harness).
__global__ void spmg_init_kernel(const float* __restrict__ kpts0,
                                 const float* __restrict__ trans,
                                 float* __restrict__ kpts0_t,
                                 unsigned long long* __restrict__ col_key,
                                 int* __restrict__ gt1_out,
                                 int B, int N0, int N1) {
  int t = blockIdx.x * blockDim.x + threadIdx.x;
  if (t < B * N1) {
    col_key[t] = ~0ull;
    gt1_out[t] = -1;
  }
  if (t < B * N0) {
    int b = t / N0;
    const float* H = trans + b * 9;
    float x = kpts0[t * 2 + 0];
    float y = kpts0[t * 2 + 1];
    float px = H[0] * x + H[1] * y + H[2];
    float py = H[3] * x + H[4] * y + H[5];
    float pz = H[6] * x + H[7] * y + H[8];
    float inv = 1.0f / (pz + 1e-8f);
    kpts0_t[t * 2 + 0] = px * inv;
    kpts0_t[t * 2 + 1] = py * inv;
  }
}

// Kernel 2: squared-distance matrix computed ENTIRELY inside
// V_WMMA_F32_16X16X4_F32 by packing the expansion
//   d^2 = (-2ax)*bx + (-2ay)*by + 1*|b|^2 + |a|^2*1
// into the K=4 dot product:
//   A row  (M=i): (-2ax, -2ay, 1, |a|^2)   lanes 0-15 = K0,K1; 16-31 = K2,K3
//   B col  (N=j): ( bx,   by, |b|^2, 1 )   lanes 0-15 = K0,K1; 16-31 = K2,K3
// All 32 lanes load branchlessly. Packed-u64 keys give min-dist-then-min-index
// argmin (matches jnp.argmin tie-breaking). Row argmin: per-lane v_min_u64
// accumulators + one 16-lane butterfly at the end. Column argmin: ds_min_u64
// into a per-workgroup LDS table (fire-and-forget, no loop-carried waits),
// flushed once per WG with global_atomic_min_u64.
__global__ __launch_bounds__(256) void spmg_dist_kernel(
    const float* __restrict__ kpts0_t,
    const float* __restrict__ kpts1,
    unsigned long long* __restrict__ row_key,
    unsigned long long* __restrict__ col_key,
    int N0, int N1) {
  extern __shared__ unsigned long long col_lds[];  // N1 entries

  const int lane = threadIdx.x & 31;
  const int wave = threadIdx.x >> 5;
  const int wgPerB = N0 / 128;
  const int b = blockIdx.x / wgPerB;
  const int m0 = (blockIdx.x % wgPerB) * 128 + wave * 16;
  const int laneSel = lane & 15;
  const bool hiHalf = lane >= 16;
  const int halfoff = (lane & 16) >> 1;  // 0 for lanes 0-15, 8 for 16-31

  // Init the per-WG column-min table.
  for (int t = threadIdx.x; t < N1; t += blockDim.x) col_lds[t] = ~0ull;
  __syncthreads();

  // ---- A operand (loop-invariant) ----
  v2f A;
  {
    int m = m0 + laneSel;
    float2 a = *(const float2*)&kpts0_t[(size_t)(b * N0 + m) * 2];
    float a2 = fmaf(a.x, a.x, a.y * a.y);
    A.x = hiHalf ? 1.0f : -2.0f * a.x;  // K2 = 1        | K0 = -2ax
    A.y = hiHalf ? a2 : -2.0f * a.y;    // K3 = |a|^2    | K1 = -2ay
  }

  unsigned long long bestRow[8];
#pragma unroll
  for (int r = 0; r < 8; ++r) bestRow[r] = ~0ull;

  const float2* __restrict__ k1base = (const float2*)kpts1 + (size_t)b * N1;

  // Double-buffered column tiles: load one tile ahead so the global_load_b64
  // latency overlaps the WMMA + min chain of the current tile.
  float2 curB = k1base[laneSel];
  for (int j0 = 0; j0 < N1; j0 += 16) {
    int jn = (j0 + 16 < N1) ? (j0 + 16) : j0;  // clamp: redundant last load
    float2 nxtB = k1base[jn + laneSel];

    float b2 = fmaf(curB.x, curB.x, curB.y * curB.y);
    v2f Bv;
    Bv.x = hiHalf ? b2 : curB.x;    // K2 = |b|^2 | K0 = bx
    Bv.y = hiHalf ? 1.0f : curB.y;  // K3 = 1     | K1 = by

    v8f c = {};
    c = __builtin_amdgcn_wmma_f32_16x16x4_f32(
        /*neg_a=*/false, A, /*neg_b=*/false, Bv,
        /*c_mod=*/(short)0, c, /*reuse_a=*/false, /*reuse_b=*/false);

    const unsigned ncol = (unsigned)(j0 + laneSel);
    unsigned long long colBest = ~0ull;
#pragma unroll
    for (int r = 0; r < 8; ++r) {
      // single-instruction clamp-to-[0,inf): v_med3_num_f32
      float d2 = __builtin_amdgcn_fmed3f(c[r], 0.0f, FLT_MAX_C);
      unsigned long long hb = ((unsigned long long)__float_as_uint(d2)) << 32;
      bestRow[r] = umin64(bestRow[r], hb | ncol);
      colBest = umin64(colBest, hb | (unsigned)(m0 + r + halfoff));
    }
    // All 32 lanes: ds_min_u64 (lanes L and L+16 hit the same slot; the LDS
    // atomic unit merges them). Fire-and-forget, no exec-mask games.
    atomicMin(&col_lds[ncol], colBest);

    curB = nxtB;
  }

  // Row reduction: lanes within each half share a row, butterfly over 16 lanes.
#pragma unroll
  for (int r = 0; r < 8; ++r) {
    unsigned long long v = bestRow[r];
    v = umin64(v, shflxor64(v, 1));
    v = umin64(v, shflxor64(v, 2));
    v = umin64(v, shflxor64(v, 4));
    v = umin64(v, shflxor64(v, 8));
    if (lane == 0) row_key[b * N0 + m0 + r] = v;
    if (lane == 16) row_key[b * N0 + m0 + r + 8] = v;
  }

  // Merge this WG's column mins into the global table (16x fewer device-scope
  // atomics than doing it per-tile).
  __syncthreads();
  for (int t = threadIdx.x; t < N1; t += blockDim.x)
    atomicMin(&col_key[b * N1 + t], col_lds[t]);
}

// Kernel 3: cross-check, threshold, emit gt_matches0 / min_dist0, scatter the
// injective inverse map into gt_matches1 (pre-initialized to -1).
__global__ void spmg_finalize_kernel(const unsigned long long* __restrict__ row_key,
                                     const unsigned long long* __restrict__ col_key,
                                     int* __restrict__ gt0,
                                     int* __restrict__ gt1,
                                     float* __restrict__ md,
                                     int B, int N0, int N1) {
  int t = blockIdx.x * blockDim.x + threadIdx.x;
  if (t >= B * N0) return;
  int b = t / N0;
  int i = t - b * N0;
  unsigned long long rk = row_key[t];
  float d2 = __uint_as_float((unsigned)(rk >> 32));
  int j = (int)(unsigned)(rk & 0xFFFFFFFFu);
  float dist = sqrtf(d2);
  int back = (int)(unsigned)(col_key[b * N1 + j] & 0xFFFFFFFFu);
  bool valid = (back == i) && (dist <= 3.0f);
  gt0[t] = valid ? j : -1;
  md[t] = dist;
  if (valid) gt1[b * N1 + j] = i;  // cross-check => at most one writer per j
}

extern "C" void kernel_launch(void* const* d_in, const int* in_sizes, int n_in,
                              void* d_out, int out_size, void* d_ws, size_t ws_size,
                              hipStream_t stream) {
  const float* kpts0 = (const float*)d_in[0];
  const float* kpts1 = (const float*)d_in[1];
  const float* trans = (const float*)d_in[2];

  const int B = in_sizes[2] / 9;
  const int N0 = in_sizes[0] / (2 * B);
  const int N1 = in_sizes[1] / (2 * B);

  // Workspace: kpts0_t | col_key (u64) | row_key (u64)  (~384 KB total)
  float* kpts0_t = (float*)d_ws;
  size_t off = ((size_t)B * N0 * 2 * sizeof(float) + 7) & ~(size_t)7;
  unsigned long long* col_key = (unsigned long long*)((char*)d_ws + off);
  unsigned long long* row_key = col_key + (size_t)B * N1;

  // Outputs flat: gt_matches0 [B*N0] i32 | gt_matches1 [B*N1] i32 | min_dist0 [B*N0] f32
  int* gt0 = (int*)d_out;
  int* gt1 = gt0 + (size_t)B * N0;
  float* md = (float*)d_out + (size_t)B * N0 + (size_t)B * N1;

  int totInit = B * (N0 > N1 ? N0 : N1);
  spmg_init_kernel<<<(totInit + 255) / 256, 256, 0, stream>>>(
      kpts0, trans, kpts0_t, col_key, gt1, B, N0, N1);

  int nBlocks = B * (N0 / 128);
  size_t ldsBytes = (size_t)N1 * sizeof(unsigned long long);  // 16 KB
  spmg_dist_kernel<<<nBlocks, 256, ldsBytes, stream>>>(kpts0_t, kpts1, row_key,
                                                       col_key, N0, N1);

  int totF = B * N0;
  spmg_finalize_kernel<<<(totF + 255) / 256, 256, 0, stream>>>(
      row_key, col_key, gt0, gt1, md, B, N0, N1);
}